// VectorQuantizer_10402410790967
// MI455X (gfx1250) — compile-verified
//
#include <hip/hip_runtime.h>
#include <hip/hip_bf16.h>

typedef __attribute__((ext_vector_type(2))) float v2f;
typedef __attribute__((ext_vector_type(8))) float v8f;

#define EDIM   256
#define NE     8192
#define TILE_K 16          // codebook entries per LDS tile
#define LDS_STRIDE 260     // 256 + 4 pad -> conflict-free ds_load_b64

// ---------------------------------------------------------------------------
// Kernel 1: codebook squared norms
// ---------------------------------------------------------------------------
__global__ __launch_bounds__(256) void vq_norms_kernel(
    const float* __restrict__ emb, float* __restrict__ norms)
{
    int k = blockIdx.x * blockDim.x + threadIdx.x;   // 0..8191
    const float4* row = (const float4*)(emb + k * EDIM);
    float s = 0.0f;
#pragma unroll 8
    for (int c = 0; c < EDIM / 4; ++c) {
        float4 v = row[c];
        s += v.x * v.x + v.y * v.y + v.z * v.z + v.w * v.w;
    }
    norms[k] = s;
}

// ---------------------------------------------------------------------------
// Kernel 2: fused GEMM + argmin using V_WMMA_F32_16X16X4_F32
//   block = 128 threads (4 waves); each wave owns 16 query rows.
//   A (queries) fully register-resident: 64 x v2f = 128 VGPRs per lane.
//   B (codebook) streamed 16 rows at a time through padded LDS.
// ---------------------------------------------------------------------------
__global__ __launch_bounds__(128, 1) void vq_argmin_kernel(
    const float* __restrict__ z, const float* __restrict__ emb,
    const float* __restrict__ norms, int* __restrict__ idx_out)
{
    __shared__ __align__(16) float lds_e[TILE_K * LDS_STRIDE];  // 16.6 KB

    const int tid  = threadIdx.x;
    const int wave = tid >> 5;
    const int lane = tid & 31;
    const int lM   = lane & 15;    // row-in-tile (A) / col-in-tile (B)
    const int half = lane >> 4;    // K sub-group per WMMA layout

    // --- query row owned by this lane (A-matrix M = lM) ---
    const int row_base = blockIdx.x * 64 + wave * 16;
    const int n  = row_base + lM;              // 0..16383
    const int b  = n >> 10;
    const int hh = (n >> 5) & 31;
    const int ww = n & 31;
    // z layout (b,c,h,w): index = b*262144 + c*1024 + hh*32 + ww
    const int zbase = b * 262144 + hh * 32 + ww;

    // --- load A fragments: a[kc] = { zf[n][K], zf[n][K+1] }, K = 4*kc + 2*half
    v2f a[64];
#pragma unroll
    for (int kc = 0; kc < 64; ++kc) {
        int K = kc * 4 + 2 * half;
        a[kc].x = z[zbase + K * 1024];
        a[kc].y = z[zbase + (K + 1) * 1024];
    }

    float bestD[8];
    int   bestK[8];
#pragma unroll
    for (int v = 0; v < 8; ++v) { bestD[v] = 3.402823466e38f; bestK[v] = 0; }

    for (int kb = 0; kb < NE; kb += TILE_K) {
        __syncthreads();   // previous tile's LDS reads complete
        // cooperative tile load: 16 rows x 256 floats as float2 (coalesced)
#pragma unroll
        for (int e = 0; e < 16; ++e) {
            int f2 = tid + e * 128;        // float2 index 0..2047
            int r  = f2 >> 7;
            int c2 = f2 & 127;
            float2 val = *(const float2*)(emb + (kb + r) * EDIM + c2 * 2);
            *(float2*)(&lds_e[r * LDS_STRIDE + c2 * 2]) = val;
        }
        if (kb + TILE_K < NE)              // hint next tile toward L2
            __builtin_prefetch(emb + (kb + TILE_K) * EDIM + lane * 64, 0, 0);
        __syncthreads();

        // 16x16 dot tile, K = 256 via 64 fp32 WMMAs
        v8f acc = {0.f, 0.f, 0.f, 0.f, 0.f, 0.f, 0.f, 0.f};
#pragma unroll
        for (int kc = 0; kc < 64; ++kc) {
            v2f bf = *(const v2f*)(&lds_e[lM * LDS_STRIDE + kc * 4 + 2 * half]);
            acc = __builtin_amdgcn_wmma_f32_16x16x4_f32(
                false, a[kc], false, bf, (short)0, acc, false, false);
        }

        // dist[M][N] = ||e_N||^2 - 2 * dot ; N = lM (column), M = v + 8*half
        float n2 = norms[kb + lM];
#pragma unroll
        for (int v = 0; v < 8; ++v) {
            float d  = n2 - 2.0f * acc[v];
            int   ki = kb + lM;
            // butterfly min across the 16 lanes holding this row's columns
#pragma unroll
            for (int off = 1; off < 16; off <<= 1) {
                float od = __shfl_xor(d, off, 32);
                int   oi = __shfl_xor(ki, off, 32);
                if (od < d || (od == d && oi < ki)) { d = od; ki = oi; }
            }
            if (d < bestD[v]) { bestD[v] = d; bestK[v] = ki; }  // strict <: keeps first min
        }
    }

    if (lM == 0) {
#pragma unroll
        for (int v = 0; v < 8; ++v)
            idx_out[row_base + v + 8 * half] = bestK[v];
    }
}

// ---------------------------------------------------------------------------
// Kernel 3: gather z_q (b,c,h,w layout) + deterministic partial loss sums
// ---------------------------------------------------------------------------
__global__ __launch_bounds__(256) void vq_gather_kernel(
    const float* __restrict__ z, const float* __restrict__ emb,
    const int* __restrict__ idx_ws, float* __restrict__ out,
    float* __restrict__ partials)
{
    __shared__ float red[256];
    int tid = threadIdx.x;
    float acc = 0.0f;
#pragma unroll
    for (int e = 0; e < 16; ++e) {
        int i = blockIdx.x * 4096 + e * 256 + tid;
        int w = i & 31, h = (i >> 5) & 31, c = (i >> 10) & 255, b = i >> 18;
        int n = (b << 10) | (h << 5) | w;
        int k = idx_ws[n];
        float q  = emb[k * EDIM + c];
        float zv = z[i];
        out[i] = q;                       // straight-through forward value
        float df = q - zv;
        acc += df * df;
    }
    red[tid] = acc;
    __syncthreads();
    for (int s = 128; s > 0; s >>= 1) {
        if (tid < s) red[tid] += red[tid + s];
        __syncthreads();
    }
    if (tid == 0) partials[blockIdx.x] = red[0];
}

// ---------------------------------------------------------------------------
// Kernel 4: final loss = (1 + beta) * mean(diff^2), fixed-order reduction
// ---------------------------------------------------------------------------
__global__ __launch_bounds__(256) void vq_loss_kernel(
    const float* __restrict__ partials, float* __restrict__ out_loss)
{
    __shared__ float red[256];
    int tid = threadIdx.x;
    float s = 0.0f;
#pragma unroll
    for (int e = 0; e < 4; ++e) s += partials[tid * 4 + e];
    red[tid] = s;
    __syncthreads();
    for (int st = 128; st > 0; st >>= 1) {
        if (tid < st) red[tid] += red[tid + st];
        __syncthreads();
    }
    if (tid == 0) out_loss[0] = 1.25f * red[0] / 4194304.0f;
}

// ---------------------------------------------------------------------------
// Kernel 5: indices to output (as float; JAX concat promotes int32->f32)
// ---------------------------------------------------------------------------
__global__ __launch_bounds__(256) void vq_idx_kernel(
    const int* __restrict__ idx_ws, float* __restrict__ out_idx)
{
    int n = blockIdx.x * 256 + threadIdx.x;
    out_idx[n] = (float)idx_ws[n];
}

// ---------------------------------------------------------------------------
extern "C" void kernel_launch(void* const* d_in, const int* in_sizes, int n_in,
                              void* d_out, int out_size, void* d_ws, size_t ws_size,
                              hipStream_t stream)
{
    const float* z   = (const float*)d_in[0];   // (16,256,32,32) f32
    const float* emb = (const float*)d_in[1];   // (8192,256) f32
    float* out = (float*)d_out;                 // [z_q_out | loss | idx]

    // workspace layout
    int*   idx_ws   = (int*)d_ws;                                   // 16384 * 4
    float* norms    = (float*)((char*)d_ws + 65536);                //  8192 * 4
    float* partials = (float*)((char*)d_ws + 65536 + 32768);        //  1024 * 4

    vq_norms_kernel <<<NE / 256, 256, 0, stream>>>(emb, norms);
    vq_argmin_kernel<<<16384 / 64, 128, 0, stream>>>(z, emb, norms, idx_ws);
    vq_gather_kernel<<<4194304 / 4096, 256, 0, stream>>>(z, emb, idx_ws, out, partials);
    vq_loss_kernel  <<<1, 256, 0, stream>>>(partials, out + 4194304);
    vq_idx_kernel   <<<16384 / 256, 256, 0, stream>>>(idx_ws, out + 4194305);
}